// GATEncoder_8400956031061
// MI455X (gfx1250) — compile-verified
//
#include <hip/hip_runtime.h>
#include <hip/hip_bf16.h>

typedef float v2f __attribute__((ext_vector_type(2)));
typedef float v8f __attribute__((ext_vector_type(8)));

#define CH 128  // feature width for both layers (4*32 and 1*128)

// ---------------------------------------------------------------------------
// Y[nRows,128] = X[nRows,128] @ W[128,128] using V_WMMA_F32_16X16X4_F32.
// One block = 256 threads = 8 waves; wave w computes the 16x16 tile at
// rows [16*blockIdx.x, +16), cols [16*w, +16). X row-panel staged in LDS.
// ---------------------------------------------------------------------------
__global__ void __launch_bounds__(256) gemm_x128(const float* __restrict__ X,
                                                 const float* __restrict__ W,
                                                 float* __restrict__ Y,
                                                 int nRows) {
  __shared__ float tileX[16 * CH];
  const int row0 = blockIdx.x * 16;
  const int t = threadIdx.x;

  // Cooperative load of the 16x128 X panel (256 thr * 8 floats = 2048 floats).
  {
    int r  = t >> 4;            // 0..15
    int c0 = (t & 15) * 8;      // 0,8,...,120
    int gr = row0 + r;
    if (gr >= nRows) gr = nRows - 1;   // clamp (keeps EXEC full for WMMA)
    const float4* xp = (const float4*)(X + (size_t)gr * CH + c0);
    float4 v0 = xp[0];
    float4 v1 = xp[1];
    *(float4*)&tileX[r * CH + c0]     = v0;
    *(float4*)&tileX[r * CH + c0 + 4] = v1;
  }
  __syncthreads();

  const int wid  = t >> 5;
  const int lane = t & 31;
  const int ml   = lane & 15;   // A: row-in-tile / B,D: col-in-tile
  const int g    = lane >> 4;   // lane-half selector
  const int n    = wid * 16 + ml;

  v8f acc = {};
#pragma unroll
  for (int ks = 0; ks < 32; ++ks) {
    const int kb = ks * 4 + 2 * g;
    v2f a;
    a.x = tileX[ml * CH + kb];
    a.y = tileX[ml * CH + kb + 1];
    v2f b;
    b.x = W[(size_t)kb * CH + n];
    b.y = W[(size_t)(kb + 1) * CH + n];
    acc = __builtin_amdgcn_wmma_f32_16x16x4_f32(false, a, false, b,
                                                (short)0, acc, false, false);
  }

#pragma unroll
  for (int r = 0; r < 8; ++r) {
    int m = row0 + r + 8 * g;       // D: vgpr r holds rows r (lanes 0-15), r+8
    if (m < nRows) Y[(size_t)m * CH + n] = acc[r];
  }
}

// ---------------------------------------------------------------------------
// alpha_s[n,h] = sum_c H[n, h*Cc + c] * aS[h*Cc + c]  (same for aD).
// One wave per node; per-head reduction via shfl_xor over groups of 32/heads.
// ---------------------------------------------------------------------------
__global__ void __launch_bounds__(256) alpha_kernel(const float* __restrict__ H,
                                                    const float* __restrict__ aS,
                                                    const float* __restrict__ aD,
                                                    float* __restrict__ outS,
                                                    float* __restrict__ outD,
                                                    int nNodes, int heads) {
  const int wid  = threadIdx.x >> 5;
  const int lane = threadIdx.x & 31;
  const int node = blockIdx.x * 8 + wid;
  if (node >= nNodes) return;

  const float4 hv = *(const float4*)(H + (size_t)node * CH + lane * 4);
  const float4 vs = *(const float4*)(aS + lane * 4);
  const float4 vd = *(const float4*)(aD + lane * 4);
  float ps = hv.x * vs.x + hv.y * vs.y + hv.z * vs.z + hv.w * vs.w;
  float pd = hv.x * vd.x + hv.y * vd.y + hv.z * vd.z + hv.w * vd.w;

  const int G = 32 / heads;   // lanes per head group
  for (int off = G >> 1; off > 0; off >>= 1) {
    ps += __shfl_xor(ps, off, 32);
    pd += __shfl_xor(pd, off, 32);
  }
  if ((lane & (G - 1)) == 0) {
    int h = lane / G;
    outS[node * heads + h] = ps;
    outD[node * heads + h] = pd;
  }
}

// ---------------------------------------------------------------------------
// Per (edge, head): ee = exp(leaky_relu(aS[src]+aD[dst])); denom[dst] += ee.
// Edge ids >= E are the implicit self-loops (src = dst = e - E).
// ---------------------------------------------------------------------------
__global__ void __launch_bounds__(256) edge_alpha_kernel(const int* __restrict__ src,
                                                         const int* __restrict__ dst,
                                                         const float* __restrict__ aS,
                                                         const float* __restrict__ aD,
                                                         float* __restrict__ ee,
                                                         float* __restrict__ denom,
                                                         int E, int Etot, int heads) {
  const int idx = blockIdx.x * blockDim.x + threadIdx.x;
  const int tot = Etot * heads;
  if (idx >= tot) return;
  const int e = idx / heads;
  const int h = idx - e * heads;
  const int s = (e < E) ? src[e] : (e - E);
  const int d = (e < E) ? dst[e] : (e - E);
  float v = aS[s * heads + h] + aD[d * heads + h];
  v = (v > 0.0f) ? v : 0.2f * v;          // leaky_relu, slope 0.2
  const float ev = expf(v);               // max-shift dropped (mathematically =)
  ee[idx] = ev;
  atomicAdd(&denom[d * heads + h], ev);
}

// ---------------------------------------------------------------------------
// One wave per edge: Agg[dst, :] += (ee/denom[dst]) * H[src, :]  (128 floats,
// float4 per lane, per-head weight).
// ---------------------------------------------------------------------------
__global__ void __launch_bounds__(256) edge_aggregate_kernel(const int* __restrict__ src,
                                                             const int* __restrict__ dst,
                                                             const float* __restrict__ ee,
                                                             const float* __restrict__ denom,
                                                             const float* __restrict__ H,
                                                             float* __restrict__ Agg,
                                                             int E, int Etot, int heads) {
  const int wid  = threadIdx.x >> 5;
  const int lane = threadIdx.x & 31;
  const int e = blockIdx.x * 8 + wid;
  if (e >= Etot) return;
  const int s = (e < E) ? src[e] : (e - E);
  const int d = (e < E) ? dst[e] : (e - E);
  const int head = (lane * heads) >> 5;   // lane / (32/heads)
  const float w = ee[e * heads + head] / (denom[d * heads + head] + 1e-16f);
  const float4 hv = *(const float4*)(H + (size_t)s * CH + lane * 4);
  float* ap = Agg + (size_t)d * CH + lane * 4;
  atomicAdd(ap + 0, w * hv.x);
  atomicAdd(ap + 1, w * hv.y);
  atomicAdd(ap + 2, w * hv.z);
  atomicAdd(ap + 3, w * hv.w);
}

// out[i] = elu(out[i] + bias[i % 128])   (layer-1 epilogue, in place)
__global__ void __launch_bounds__(256) bias_elu_kernel(float* __restrict__ A,
                                                       const float* __restrict__ bias,
                                                       int total) {
  const int i = blockIdx.x * blockDim.x + threadIdx.x;
  if (i >= total) return;
  float v = A[i] + bias[i & (CH - 1)];
  A[i] = (v > 0.0f) ? v : expm1f(v);
  }

// out[i] += bias[i % 128]   (final epilogue)
__global__ void __launch_bounds__(256) bias_add_kernel(float* __restrict__ A,
                                                       const float* __restrict__ bias,
                                                       int total) {
  const int i = blockIdx.x * blockDim.x + threadIdx.x;
  if (i >= total) return;
  A[i] += bias[i & (CH - 1)];
}

// ---------------------------------------------------------------------------
extern "C" void kernel_launch(void* const* d_in, const int* in_sizes, int n_in,
                              void* d_out, int out_size, void* d_ws, size_t ws_size,
                              hipStream_t stream) {
  const float* x   = (const float*)d_in[0];
  const int*   ei  = (const int*)  d_in[1];
  const float* W1  = (const float*)d_in[2];
  const float* as1 = (const float*)d_in[3];
  const float* ad1 = (const float*)d_in[4];
  const float* b1  = (const float*)d_in[5];
  const float* W2  = (const float*)d_in[6];
  const float* as2 = (const float*)d_in[7];
  const float* ad2 = (const float*)d_in[8];
  const float* b2  = (const float*)d_in[9];

  const int nNodes = in_sizes[0] / CH;
  const int E      = in_sizes[1] / 2;
  const int Etot   = E + nNodes;
  const int* src = ei;
  const int* dst = ei + E;
  float* out = (float*)d_out;

  // Workspace carve-up (floats). h2 reuses h1; hmid computed in place in agg1.
  float* ws = (float*)d_ws;
  size_t off = 0;
  float* h1   = ws + off; off += (size_t)nNodes * CH;
  float* agg1 = ws + off; off += (size_t)nNodes * CH;
  float* asn  = ws + off; off += (size_t)nNodes * 4;
  float* adn  = ws + off; off += (size_t)nNodes * 4;
  float* den  = ws + off; off += (size_t)nNodes * 4;
  float* ee   = ws + off; off += (size_t)Etot * 4;
  (void)ws_size; (void)n_in; (void)out_size;

  hipMemsetAsync(agg1, 0, (size_t)nNodes * CH * sizeof(float), stream);
  hipMemsetAsync(den,  0, (size_t)nNodes * 4 * sizeof(float), stream);
  hipMemsetAsync(d_out, 0, (size_t)nNodes * CH * sizeof(float), stream);

  const int rowBlocks = (nNodes + 15) / 16;

  // ---- Layer 1: heads=4, C=32, concat ----
  gemm_x128<<<rowBlocks, 256, 0, stream>>>(x, W1, h1, nNodes);
  alpha_kernel<<<(nNodes + 7) / 8, 256, 0, stream>>>(h1, as1, ad1, asn, adn, nNodes, 4);
  edge_alpha_kernel<<<(Etot * 4 + 255) / 256, 256, 0, stream>>>(src, dst, asn, adn, ee, den, E, Etot, 4);
  edge_aggregate_kernel<<<(Etot + 7) / 8, 256, 0, stream>>>(src, dst, ee, den, h1, agg1, E, Etot, 4);
  bias_elu_kernel<<<(nNodes * CH + 255) / 256, 256, 0, stream>>>(agg1, b1, nNodes * CH);

  // ---- Layer 2: heads=1, C=128, mean(=identity) ----
  gemm_x128<<<rowBlocks, 256, 0, stream>>>(agg1, W2, h1, nNodes);   // h2 -> h1
  hipMemsetAsync(den, 0, (size_t)nNodes * sizeof(float), stream);
  alpha_kernel<<<(nNodes + 7) / 8, 256, 0, stream>>>(h1, as2, ad2, asn, adn, nNodes, 1);
  edge_alpha_kernel<<<(Etot + 255) / 256, 256, 0, stream>>>(src, dst, asn, adn, ee, den, E, Etot, 1);
  edge_aggregate_kernel<<<(Etot + 7) / 8, 256, 0, stream>>>(src, dst, ee, den, h1, out, E, Etot, 1);
  bias_add_kernel<<<(nNodes * CH + 255) / 256, 256, 0, stream>>>(out, b2, nNodes * CH);
}